// RGBCLAHEOperator_63797444215449
// MI455X (gfx1250) — compile-verified
//
#include <hip/hip_runtime.h>
#include <hip/hip_bf16.h>

// ---------------------------------------------------------------------------
// CLAHE pipeline for MI455X (gfx1250, wave32).
//  Pass 1: RGB -> HLS (u8, packed stores) + per-tile 256-bin histogram in LDS
//  Pass 2: per-tile clip/redistribute + CDF via V_WMMA_F32_16X16X4_F32 -> LUT
//  Pass 3: bilinear LUT blend + HLS -> RGB (float4 stores, bandwidth bound)
// ---------------------------------------------------------------------------

#define IMG_B   32
#define IMG_H   512
#define IMG_W   512
#define TILES   8
#define TDIM    64            // tile is 64x64 pixels
#define NBINS   256
#define NTILES  (IMG_B * TILES * TILES)          // 2048
#define NPIX    ((size_t)IMG_B * IMG_H * IMG_W)  // 8388608
#define CLIPV   640           // max(int(40.0 * 4096 / 256), 1)
#define AREA_F  4096.0f

typedef float v2f __attribute__((ext_vector_type(2)));
typedef float v8f __attribute__((ext_vector_type(8)));

static __device__ __forceinline__ v8f wmma4(v2f a, v2f b, v8f c) {
  // D = A(16x4,f32) x B(4x16,f32) + C(16x16,f32)
  return __builtin_amdgcn_wmma_f32_16x16x4_f32(
      /*neg_a=*/false, a, /*neg_b=*/false, b,
      /*c_mod=*/(short)0, c, /*reuse_a=*/false, /*reuse_b=*/false);
}

static __device__ __forceinline__ float clamp255(float v) {
  return fminf(fmaxf(v, 0.0f), 255.0f);
}

// ---------------------------------------------------------------------------
// Pass 1: one block per 64x64 tile. 256 threads x 16 pixels (4x float4).
// ---------------------------------------------------------------------------
__global__ __launch_bounds__(256)
void rgb2hls_hist_kernel(const float* __restrict__ img,
                         unsigned char* __restrict__ H8o,
                         unsigned char* __restrict__ S8o,
                         unsigned char* __restrict__ L8o,
                         int* __restrict__ hist) {
  __shared__ int lh[NBINS];
  const int tid = threadIdx.x;
  lh[tid] = 0;
  __syncthreads();

  const int blk = blockIdx.x;            // global tile id: b*64 + ty*8 + tx
  const int b   = blk >> 6;
  const int t   = blk & 63;
  const int ty  = t >> 3;
  const int tx  = t & 7;
  const size_t plane = (size_t)IMG_H * IMG_W;
  const float* base  = img + (size_t)b * 3 * plane;

#pragma unroll
  for (int i = 0; i < 4; ++i) {
    const int p  = tid * 4 + i * 1024;   // 4 consecutive pixels, same tile row
    const int ly = p >> 6;
    const int lx = p & 63;
    const int y  = ty * TDIM + ly;
    const int x  = tx * TDIM + lx;
    const size_t pix = (size_t)y * IMG_W + x;

    if (i + 1 < 4) {  // gfx1250 global_prefetch_b8
      const int pn = tid * 4 + (i + 1) * 1024;
      const size_t npix = (size_t)(ty * TDIM + (pn >> 6)) * IMG_W +
                          tx * TDIM + (pn & 63);
      __builtin_prefetch(base + npix, 0, 3);
    }

    const float4 r4 = *(const float4*)(base + pix);
    const float4 g4 = *(const float4*)(base + pix + plane);
    const float4 b4 = *(const float4*)(base + pix + 2 * plane);
    const float rr[4] = {r4.x, r4.y, r4.z, r4.w};
    const float gg[4] = {g4.x, g4.y, g4.z, g4.w};
    const float bb[4] = {b4.x, b4.y, b4.z, b4.w};

    unsigned int hp = 0, sp = 0, lp = 0;
#pragma unroll
    for (int j = 0; j < 4; ++j) {
      const float r  = clamp255(floorf(rr[j])) * (1.0f / 255.0f);
      const float g  = clamp255(floorf(gg[j])) * (1.0f / 255.0f);
      const float bl = clamp255(floorf(bb[j])) * (1.0f / 255.0f);

      const float vmax = fmaxf(fmaxf(r, g), bl);
      const float vmin = fminf(fminf(r, g), bl);
      const float l    = (vmax + vmin) * 0.5f;
      const float diff = vmax - vmin;
      const float denom_s = (l < 0.5f) ? (vmax + vmin) : (2.0f - vmax - vmin);
      const float s = (diff > 0.0f) ? diff / fmaxf(denom_s, 1e-12f) : 0.0f;
      const float safe = fmaxf(diff, 1e-12f);
      float h;
      if (vmax == r)      h = 60.0f * (g - bl) / safe;
      else if (vmax == g) h = 120.0f + 60.0f * (bl - r) / safe;
      else                h = 240.0f + 60.0f * (r - g) / safe;
      h = (diff > 0.0f) ? h : 0.0f;
      if (h < 0.0f) h += 360.0f;

      const unsigned int h8 = (unsigned int)clamp255(rintf(h * 0.5f));
      const unsigned int l8 = (unsigned int)clamp255(rintf(l * 255.0f));
      const unsigned int s8 = (unsigned int)clamp255(rintf(s * 255.0f));
      hp |= h8 << (8 * j);
      sp |= s8 << (8 * j);
      lp |= l8 << (8 * j);
      atomicAdd(&lh[l8], 1);
    }

    const size_t gp = (size_t)b * plane + pix;   // multiple of 4
    *(unsigned int*)(H8o + gp) = hp;
    *(unsigned int*)(S8o + gp) = sp;
    *(unsigned int*)(L8o + gp) = lp;
  }
  __syncthreads();
  hist[(size_t)blk * NBINS + tid] = lh[tid];
}

// ---------------------------------------------------------------------------
// Pass 2: one wave32 per tile (8 waves / block). Clip + redistribute, then
// cumsum of 256 bins via 12x V_WMMA_F32_16X16X4_F32:
//   cdf = M*U + Lstrict*(M*Ones)   (M is hist as 16x16 row-major)
// All register-array indices are compile-time -> no scratch spills.
// ---------------------------------------------------------------------------
__global__ __launch_bounds__(256)
void clahe_lut_kernel(const int* __restrict__ hist,
                      unsigned char* __restrict__ lut) {
  __shared__ float fh[8 * NBINS];
  const int tid  = threadIdx.x;
  const int wave = tid >> 5;
  const int lane = tid & 31;
  const int tile = blockIdx.x * 8 + wave;
  const int half = lane >> 4;       // 0: lanes 0-15, 1: lanes 16-31
  const int ln   = lane & 15;       // A-layout row / B- & C/D-layout column

  // Load this wave's 256 bins (coalesced) and clip/redistribute in registers.
  const int* hg = hist + (size_t)tile * NBINS;
  int hv[8];
  int ex = 0;
#pragma unroll
  for (int j = 0; j < 8; ++j) {
    hv[j] = hg[lane + 32 * j];
    ex += max(hv[j] - CLIPV, 0);
  }
#pragma unroll
  for (int m = 16; m >= 1; m >>= 1) ex += __shfl_xor(ex, m, 32);

  const int addall   = ex >> 8;         // excess // 256
  const int residual = ex & 255;        // excess %  256
  const int step     = (residual > 0) ? max(NBINS / residual, 1) : NBINS;

  float* f = fh + wave * NBINS;
#pragma unroll
  for (int j = 0; j < 8; ++j) {
    const int bin = lane + 32 * j;
    int v = min(hv[j], CLIPV) + addall;
    if (residual > 0 && (bin % step == 0) && (bin / step < residual)) v += 1;
    f[bin] = (float)v;
  }
  __syncthreads();   // uniform: all 256 threads reach here

  // A-layout chunks of M: VGPR0 = local K=2h, VGPR1 = local K=2h+1; row = ln.
  v2f a[4];
#pragma unroll
  for (int k = 0; k < 4; ++k) {
    const int c0 = 4 * k + 2 * half;
    v2f av = { f[ln * 16 + c0], f[ln * 16 + c0 + 1] };
    a[k] = av;
  }

  // P = M x U, U[K][c] = (K <= c). B-layout: lane col = ln, rows K0, K0+1.
  v8f P = {0.f, 0.f, 0.f, 0.f, 0.f, 0.f, 0.f, 0.f};
#pragma unroll
  for (int k = 0; k < 4; ++k) {
    const int K0 = 4 * k + 2 * half;
    v2f bu = { (K0     <= ln) ? 1.0f : 0.0f,
               (K0 + 1 <= ln) ? 1.0f : 0.0f };
    P = wmma4(a[k], bu, P);
  }

  // T = M x Ones : T[r][c] = rowsum[r] (C/D layout: lane holds rows v+8*half)
  v8f T = {0.f, 0.f, 0.f, 0.f, 0.f, 0.f, 0.f, 0.f};
  v2f ones2 = {1.0f, 1.0f};
#pragma unroll
  for (int k = 0; k < 4; ++k) T = wmma4(a[k], ones2, T);

  float tv[8];
#pragma unroll
  for (int v = 0; v < 8; ++v) tv[v] = T[v];   // rowsum[v + 8*half]

  // Cross-half exchange: send exactly what the partner half needs (4 shuffles).
  //  half-0 lanes must receive rowsum[8,9,12,13]  (partner tv[0,1,4,5])
  //  half-1 lanes must receive rowsum[2,3,6,7]    (partner tv[2,3,6,7])
  const float s0 = __shfl_xor(half ? tv[0] : tv[2], 16, 32);
  const float s1 = __shfl_xor(half ? tv[1] : tv[3], 16, 32);
  const float s2 = __shfl_xor(half ? tv[4] : tv[6], 16, 32);
  const float s3 = __shfl_xor(half ? tv[5] : tv[7], 16, 32);

  // B-layout operands of T per chunk k: rows {4k+2h, 4k+2h+1} (const indices).
  float tb0[4], tb1[4];
  tb0[0] = half ? s0    : tv[0];  tb1[0] = half ? s1    : tv[1];
  tb0[1] = half ? s2    : tv[4];  tb1[1] = half ? s3    : tv[5];
  tb0[2] = half ? tv[2] : s0;     tb1[2] = half ? tv[3] : s1;
  tb0[3] = half ? tv[6] : s2;     tb1[3] = half ? tv[7] : s3;

  // cdf = Lstrict x T_b + P.  A-layout Lstrict[r][K] = (K < r).
  v8f D = P;
#pragma unroll
  for (int k = 0; k < 4; ++k) {
    const int K0 = 4 * k + 2 * half;
    v2f la = { (K0     < ln) ? 1.0f : 0.0f,
               (K0 + 1 < ln) ? 1.0f : 0.0f };
    v2f tb = { tb0[k], tb1[k] };
    D = wmma4(la, tb, D);
  }

  // LUT: lut[bin] = clip(round(cdf * 255/4096), 0, 255)
  const float scale = 255.0f / AREA_F;
  unsigned char* lo = lut + (size_t)tile * NBINS;
#pragma unroll
  for (int v = 0; v < 8; ++v) {
    const int bin = (v + 8 * half) * 16 + ln;   // C/D: row=v+8h, col=ln
    lo[bin] = (unsigned char)clamp255(rintf(D[v] * scale));
  }
}

// ---------------------------------------------------------------------------
// Pass 3: bilinear LUT blend + HLS -> RGB. 4 consecutive pixels per thread:
// packed u8 loads (b32) and float4 (b128) stores per plane.
// ---------------------------------------------------------------------------
__global__ __launch_bounds__(256)
void clahe_apply_kernel(const unsigned char* __restrict__ H8i,
                        const unsigned char* __restrict__ S8i,
                        const unsigned char* __restrict__ L8i,
                        const unsigned char* __restrict__ lut,
                        float* __restrict__ out) {
  const size_t plane = (size_t)IMG_H * IMG_W;            // 2^18
  const size_t gp0 = (size_t)blockIdx.x * 1024 + (size_t)threadIdx.x * 4;
  const int b   = (int)(gp0 >> 18);
  const int rem = (int)(gp0 & (plane - 1));
  const int y  = rem >> 9;            // / 512
  const int x0 = rem & 511;           // multiple of 4; 4 pixels share row y

  const unsigned int h4 = *(const unsigned int*)(H8i + gp0);
  const unsigned int s4 = *(const unsigned int*)(S8i + gp0);
  const unsigned int l4 = *(const unsigned int*)(L8i + gp0);

  // Vertical interpolation terms are shared by the 4 pixels.
  const float tyf = (float)y * (1.0f / TDIM) - 0.5f;
  int ty1 = (int)floorf(tyf);
  const float ya = tyf - (float)ty1;
  const int ty2 = min(ty1 + 1, TILES - 1);
  ty1 = max(ty1, 0);

  const unsigned char* lb = lut + (size_t)b * TILES * TILES * NBINS;
  __builtin_prefetch(lb + (ty1 * TILES + (x0 >> 6)) * NBINS, 0, 3);

  float ro[4], go[4], bo[4];
#pragma unroll
  for (int j = 0; j < 4; ++j) {
    const int x = x0 + j;
    const int v = (int)((l4 >> (8 * j)) & 255u);

    const float txf = (float)x * (1.0f / TDIM) - 0.5f;
    int tx1 = (int)floorf(txf);
    const float xa = txf - (float)tx1;
    const int tx2 = min(tx1 + 1, TILES - 1);
    tx1 = max(tx1, 0);

    const float g11 = (float)lb[(ty1 * TILES + tx1) * NBINS + v];
    const float g12 = (float)lb[(ty1 * TILES + tx2) * NBINS + v];
    const float g21 = (float)lb[(ty2 * TILES + tx1) * NBINS + v];
    const float g22 = (float)lb[(ty2 * TILES + tx2) * NBINS + v];

    const float res = g11 * (1.0f - xa) * (1.0f - ya) +
                      g12 * xa * (1.0f - ya) +
                      g21 * (1.0f - xa) * ya +
                      g22 * xa * ya;
    const float lnew = clamp255(rintf(res));

    // HLS -> RGB
    const float h = (float)((h4 >> (8 * j)) & 255u) * 2.0f;
    const float l = lnew * (1.0f / 255.0f);
    const float s = (float)((s4 >> (8 * j)) & 255u) * (1.0f / 255.0f);
    const float p2 = (l <= 0.5f) ? l * (1.0f + s) : (l + s - l * s);
    const float p1 = 2.0f * l - p2;

    auto hue = [p1, p2](float hh) -> float {
      hh = fmodf(hh, 360.0f);
      if (hh < 0.0f) hh += 360.0f;
      hh *= (1.0f / 60.0f);
      if (hh < 1.0f) return p1 + (p2 - p1) * hh;
      if (hh < 3.0f) return p2;
      if (hh < 4.0f) return p1 + (p2 - p1) * (4.0f - hh);
      return p1;
    };

    float r  = hue(h + 120.0f);
    float g  = hue(h);
    float bl = hue(h - 120.0f);
    if (!(s > 0.0f)) { r = l; g = l; bl = l; }

    ro[j] = clamp255(rintf(r  * 255.0f));
    go[j] = clamp255(rintf(g  * 255.0f));
    bo[j] = clamp255(rintf(bl * 255.0f));
  }

  float* ob = out + (size_t)b * 3 * plane + (size_t)y * IMG_W + x0;  // 16B aligned
  *(float4*)(ob)             = make_float4(ro[0], ro[1], ro[2], ro[3]);
  *(float4*)(ob + plane)     = make_float4(go[0], go[1], go[2], go[3]);
  *(float4*)(ob + 2 * plane) = make_float4(bo[0], bo[1], bo[2], bo[3]);
}

// ---------------------------------------------------------------------------
extern "C" void kernel_launch(void* const* d_in, const int* in_sizes, int n_in,
                              void* d_out, int out_size, void* d_ws, size_t ws_size,
                              hipStream_t stream) {
  const float* img = (const float*)d_in[0];
  float* out = (float*)d_out;
  unsigned char* ws = (unsigned char*)d_ws;

  // Workspace layout (~26.5 MB):
  unsigned char* H8 = ws;                              // NPIX bytes
  unsigned char* S8 = ws + NPIX;                       // NPIX bytes
  unsigned char* L8 = ws + 2 * NPIX;                   // NPIX bytes
  int*           hist = (int*)(ws + 3 * NPIX);         // NTILES*256*4 bytes
  unsigned char* lut  = ws + 3 * NPIX +
                        (size_t)NTILES * NBINS * sizeof(int);  // NTILES*256 bytes

  rgb2hls_hist_kernel<<<NTILES, 256, 0, stream>>>(img, H8, S8, L8, hist);
  clahe_lut_kernel<<<NTILES / 8, 256, 0, stream>>>(hist, lut);
  clahe_apply_kernel<<<(unsigned)(NPIX / 1024), 256, 0, stream>>>(H8, S8, L8, lut, out);
}